// GRUModel_15393162789002
// MI455X (gfx1250) — compile-verified
//
#include <hip/hip_runtime.h>
#include <hip/hip_bf16.h>

// Problem constants (from the reference): T=4096, B=256, I=2, H=32, O=1.
// Key algebraic fact: output depends ONLY on batch element b = B-1 = 255.

#define T_LEN 4096
#define BATCH 256
#define HID   32
#define G3    96   // 3*HID

typedef __attribute__((ext_vector_type(2))) float v2f;
typedef __attribute__((ext_vector_type(8))) float v8f;

// ---------------------------------------------------------------------------
// Kernel 1: input projection for batch 255 only, via V_WMMA_F32_16X16X4_F32.
//   xg[t][g] = x[t,255,0]*w_ih[g,0] + x[t,255,1]*w_ih[g,1] + b_ih[g]
// Tile: 16 t-rows x 16 g-cols. K padded 2->4; only the K=0 slot (VGPR0,
// lanes 0-15) is populated per WMMA — unambiguous under any B layout — and
// the two input channels are accumulated with two chained WMMAs.
// ---------------------------------------------------------------------------
__global__ __launch_bounds__(32)
void gru_xg_wmma(const float* __restrict__ x, const float* __restrict__ w_ih,
                 const float* __restrict__ b_ih, float* __restrict__ xg)
{
    const int lane  = threadIdx.x & 31;
    const int half  = lane >> 4;    // 0: rows M=0..7 / K src, 1: rows M=8..15
    const int l15   = lane & 15;
    const int tbase = blockIdx.x * 16;
    const int gbase = blockIdx.y * 16;
    const int g     = gbase + l15;

    const float bias = b_ih[g];

#if __has_builtin(__builtin_amdgcn_wmma_f32_16x16x4_f32)
    // A (16x4 f32): VGPR0 = K0 (lanes 0-15) / K2 (lanes 16-31); VGPR1 = K1/K3.
    // We use only K0 and zero everything else; lanes 16-31 supply zeros.
    const int    trow = tbase + l15;
    const float2 xv = *(const float2*)(x + ((size_t)trow * BATCH + (BATCH - 1)) * 2);
    const float2 wv = *(const float2*)(w_ih + 2 * g);
    const float x0 = half ? 0.f : xv.x;
    const float x1 = half ? 0.f : xv.y;
    const float w0 = half ? 0.f : wv.x;
    const float w1 = half ? 0.f : wv.y;

    v8f c;
    #pragma unroll
    for (int v = 0; v < 8; ++v) c[v] = bias;   // fold b_ih into accumulator

    const v2f aK0 = {x0, 0.f}, bK0 = {w0, 0.f};   // channel i=0 at K=0
    const v2f aK1 = {x1, 0.f}, bK1 = {w1, 0.f};   // channel i=1 at K=0
    c = __builtin_amdgcn_wmma_f32_16x16x4_f32(false, aK0, false, bK0,
                                              (short)0, c, false, false);
    c = __builtin_amdgcn_wmma_f32_16x16x4_f32(false, aK1, false, bK1,
                                              (short)0, c, false, false);

    // D layout: VGPR v -> row tbase+v (lanes 0-15) / tbase+8+v (lanes 16-31),
    // col = gbase + (lane mod 16).
    const int row0 = tbase + half * 8;
    #pragma unroll
    for (int v = 0; v < 8; ++v)
        xg[(size_t)(row0 + v) * G3 + g] = c[v];
#else
    // Scalar fallback (same output layout).
    const float2 wv = *(const float2*)(w_ih + 2 * g);
    const int row0 = tbase + half * 8;
    #pragma unroll
    for (int v = 0; v < 8; ++v) {
        const float2 xv = *(const float2*)(x + ((size_t)(row0 + v) * BATCH + (BATCH - 1)) * 2);
        xg[(size_t)(row0 + v) * G3 + g] = xv.x * wv.x + xv.y * wv.y + bias;
    }
#endif
}

// ---------------------------------------------------------------------------
// Kernel 2: the serial recurrence — one wave32, lane L owns hidden unit L.
// w_hh rows L, L+32, L+64 live in 96 VGPRs/lane; h broadcast via LDS;
// fc head fused via shfl_xor reduction. Strictly serial: 4096 steps.
// ---------------------------------------------------------------------------
__device__ __forceinline__ float fast_sigmoid(float v) {
    return 1.0f / (1.0f + __expf(-v));
}
__device__ __forceinline__ float fast_tanh(float v) {
    const float e = __expf(-2.0f * v);
    return (1.0f - e) / (1.0f + e);
}

__global__ __launch_bounds__(32)
void gru_recurrence(const float* __restrict__ xg, const float* __restrict__ w_hh,
                    const float* __restrict__ b_hh, const float* __restrict__ w_fc,
                    const float* __restrict__ b_fc, float* __restrict__ out)
{
    __shared__ alignas(16) float hsh[HID];   // 16B-aligned: read as ds_load_b128
    const int L = threadIdx.x & 31;

    // Persistent per-lane weights: rows L (r), L+32 (z), L+64 (n) of w_hh.
    float wr[HID], wz[HID], wn[HID];
    {
        const float4* pr = (const float4*)(w_hh + (size_t)L * HID);
        const float4* pz = (const float4*)(w_hh + (size_t)(L + 32) * HID);
        const float4* pn = (const float4*)(w_hh + (size_t)(L + 64) * HID);
        #pragma unroll
        for (int q = 0; q < 8; ++q) {
            const float4 a = pr[q], b = pz[q], c = pn[q];
            wr[4*q+0]=a.x; wr[4*q+1]=a.y; wr[4*q+2]=a.z; wr[4*q+3]=a.w;
            wz[4*q+0]=b.x; wz[4*q+1]=b.y; wz[4*q+2]=b.z; wz[4*q+3]=b.w;
            wn[4*q+0]=c.x; wn[4*q+1]=c.y; wn[4*q+2]=c.z; wn[4*q+3]=c.w;
        }
    }
    const float br = b_hh[L], bz = b_hh[L + 32], bn = b_hh[L + 64];
    const float wo = w_fc[L], bo = b_fc[0];

    float h[HID];
    #pragma unroll
    for (int k = 0; k < HID; ++k) h[k] = 0.0f;
    float hself = 0.0f;

    // Software pipeline: xg for step t is loaded one iteration ahead.
    float xr = xg[L], xz = xg[32 + L], xn = xg[64 + L];

    for (int t = 0; t < T_LEN; ++t) {
        // Prefetch next step's xg (clamped; hides L2 latency under the FMAs).
        const int tn = (t + 1 < T_LEN) ? (t + 1) : (T_LEN - 1);
        const float* np = xg + (size_t)tn * G3 + L;
        const float nxr = np[0], nxz = np[32], nxn = np[64];

        // hg = w_hh @ h + b_hh for this lane's three gate rows.
        float ar = br, az = bz, an = bn;
        #pragma unroll
        for (int k = 0; k < HID; ++k) {
            ar = fmaf(wr[k], h[k], ar);
            az = fmaf(wz[k], h[k], az);
            an = fmaf(wn[k], h[k], an);
        }

        const float r  = fast_sigmoid(xr + ar);
        const float z  = fast_sigmoid(xz + az);
        const float n  = fast_tanh(xn + r * an);
        const float hn = (1.0f - z) * n + z * hself;
        hself = hn;

        // Fused fc head: out[t] = sum_L hn*w_fc[L] + b_fc.
        float po = hn * wo;
        #pragma unroll
        for (int off = 16; off > 0; off >>= 1)
            po += __shfl_xor(po, off, 32);
        if (L == 0) out[t] = po + bo;

        // Broadcast new h to all lanes via LDS (in-order within the wave).
        hsh[L] = hn;
        __syncthreads();
        #pragma unroll
        for (int q = 0; q < 8; ++q) {
            const float4 v = ((const float4*)hsh)[q];
            h[4*q+0]=v.x; h[4*q+1]=v.y; h[4*q+2]=v.z; h[4*q+3]=v.w;
        }
        __syncthreads();

        xr = nxr; xz = nxz; xn = nxn;
    }
}

extern "C" void kernel_launch(void* const* d_in, const int* in_sizes, int n_in,
                              void* d_out, int out_size, void* d_ws, size_t ws_size,
                              hipStream_t stream) {
    const float* x    = (const float*)d_in[0];   // (4096, 256, 2)
    const float* w_ih = (const float*)d_in[1];   // (96, 2)
    const float* w_hh = (const float*)d_in[2];   // (96, 32)
    const float* b_ih = (const float*)d_in[3];   // (96,)
    const float* b_hh = (const float*)d_in[4];   // (96,)
    const float* w_fc = (const float*)d_in[5];   // (1, 32)
    const float* b_fc = (const float*)d_in[6];   // (1,)
    float* out = (float*)d_out;                  // (4096, 1)
    float* xg  = (float*)d_ws;                   // 4096*96 floats = 1.5 MB

    // Kernel 1: WMMA input projection for batch 255 -> xg workspace.
    dim3 grid1(T_LEN / 16, G3 / 16);
    gru_xg_wmma<<<grid1, 32, 0, stream>>>(x, w_ih, b_ih, xg);

    // Kernel 2: serial GRU recurrence + fused fc head (one wave32).
    gru_recurrence<<<1, 32, 0, stream>>>(xg, w_hh, b_hh, w_fc, b_fc, out);
}